// KMeansMemory_39891656245788
// MI455X (gfx1250) — compile-verified
//
#include <hip/hip_runtime.h>
#include <stdint.h>

// Problem constants (from the reference): visual_features (4, 32, 256, 768) f32
#define B_ 4
#define T_ 32
#define N_ 256
#define D_ 768
#define K_ 1024          // BUFFER_SIZE
#define M_ 1280          // K_ + N_ rows of "data" per frame
#define NFRAMES 28       // T_ - num_init (num_init = 4)
#define NITERS 2

typedef __attribute__((ext_vector_type(2))) float v2f;
typedef __attribute__((ext_vector_type(8))) float v8f;

// ---------------------------------------------------------------------------
// init: centers[b] = first 1024 tokens (frames 0..3 are contiguous), counts=1
// ---------------------------------------------------------------------------
__global__ void km_init_kernel(const float* __restrict__ vf,
                               float* __restrict__ centers,
                               float* __restrict__ counts) {
    int b = blockIdx.y, k = blockIdx.x;                    // grid (1024, 4)
    const float4* src = (const float4*)(vf + ((size_t)b * T_ * N_ + k) * D_);
    float4* dst = (float4*)(centers + ((size_t)b * K_ + k) * D_);
    for (int d = threadIdx.x; d < D_ / 4; d += blockDim.x) dst[d] = src[d];
    if (threadIdx.x == 0) counts[b * K_ + k] = 1.0f;
}

// ---------------------------------------------------------------------------
// build_data: freeze data = [centers ; frame tokens], weights = [counts ; 1]
// ---------------------------------------------------------------------------
__global__ void km_build_data_kernel(const float* __restrict__ vf,
                                     const float* __restrict__ centers,
                                     const float* __restrict__ counts,
                                     float* __restrict__ data,
                                     float* __restrict__ weights,
                                     int frame) {
    int b = blockIdx.y, m = blockIdx.x;                    // grid (1280, 4)
    float4* dst = (float4*)(data + ((size_t)b * M_ + m) * D_);
    const float4* src;
    if (m < K_) {
        src = (const float4*)(centers + ((size_t)b * K_ + m) * D_);
    } else {
        int t = 4 + frame, n = m - K_;
        src = (const float4*)(vf + (((size_t)b * T_ + t) * N_ + n) * D_);
    }
    for (int d = threadIdx.x; d < D_ / 4; d += blockDim.x) dst[d] = src[d];
    if (threadIdx.x == 0)
        weights[b * M_ + m] = (m < K_) ? counts[b * K_ + m] : 1.0f;
}

// ---------------------------------------------------------------------------
// prep_iter: assign keys = max, zero accumulators, compute ||c_k||^2
// ---------------------------------------------------------------------------
__global__ __launch_bounds__(256)
void km_prep_iter_kernel(const float* __restrict__ centers,
                         float* __restrict__ acc,
                         float* __restrict__ acc_counts,
                         float* __restrict__ cnorm,
                         unsigned long long* __restrict__ assign) {
    int b = blockIdx.y, m = blockIdx.x;                    // grid (1280, 4)
    if (threadIdx.x == 0) assign[b * M_ + m] = ~0ull;
    if (m >= K_) return;                                   // uniform per block
    int k = m;
    const float* c = centers + ((size_t)b * K_ + k) * D_;
    float* a = acc + ((size_t)b * K_ + k) * D_;
    float p = 0.f;
    for (int d = threadIdx.x; d < D_; d += blockDim.x) {
        float v = c[d];
        p += v * v;
        a[d] = 0.f;
    }
    __shared__ float red[256];
    red[threadIdx.x] = p;
    __syncthreads();
    for (int s = 128; s > 0; s >>= 1) {
        if (threadIdx.x < s) red[threadIdx.x] += red[threadIdx.x + s];
        __syncthreads();
    }
    if (threadIdx.x == 0) {
        cnorm[b * K_ + k] = red[0];
        acc_counts[b * K_ + k] = 0.f;
    }
}

// ---------------------------------------------------------------------------
// score + fused argmin: S(m,k) = ||c_k||^2 - 2 x_m . c_k via fp32 WMMA.
// Workgroup tile: 64 m-rows x 128 centers; each of 8 waves owns a 32x32
// output block (2x2 register blocking -> 1 fragment load per WMMA).
// LDS double-buffered depth-32 chunks filled by GLOBAL_LOAD_ASYNC_TO_LDS_B128
// (ASYNCcnt): zero VGPR staging cost, transfers hide under the WMMA loop.
// ---------------------------------------------------------------------------
#define TM 64
#define TKK 128
#define DC 32
#define LDSP 36   // 36*lr mod 64 distinct for lr<16 -> conflict-free gathers

__device__ __forceinline__ unsigned int fkey(float f) {
    // monotonic float -> u32 (min over keys == min over floats)
    unsigned int u = __float_as_uint(f);
    return (u & 0x80000000u) ? ~u : (u | 0x80000000u);
}

__device__ __forceinline__ unsigned long long shfl_xor_u64_16(unsigned long long x, int mask) {
    unsigned int lo = (unsigned int)x;
    unsigned int hi = (unsigned int)(x >> 32);
    lo = __shfl_xor(lo, mask, 16);   // stays within 16-lane half-wave
    hi = __shfl_xor(hi, mask, 16);
    return ((unsigned long long)hi << 32) | lo;
}

// async DMA: copies 16 bytes from the global address sbase + voff to the LDS address in lds_off, tracked by ASYNCcnt
__device__ __forceinline__ void async_b128(unsigned int lds_off, unsigned int voff,
                                           const float* sbase) {
    asm volatile("global_load_async_to_lds_b128 %0, %1, %2"
                 :: "v"(lds_off), "v"(voff), "s"(sbase) : "memory");
}
__device__ __forceinline__ void wait_async0() {
    asm volatile("s_wait_asynccnt 0x0" ::: "memory");
}

__global__ __launch_bounds__(256)
void km_score_argmin_kernel(const float* __restrict__ data,
                            const float* __restrict__ centers,
                            const float* __restrict__ cnorm,
                            unsigned long long* __restrict__ assign) {
    int b  = blockIdx.z;
    int m0 = blockIdx.x * TM;          // data-row tile
    int k0 = blockIdx.y * TKK;         // center tile

    __shared__ float lds_a[2][TM][LDSP];    // 2*64*36*4  = 18432 B
    __shared__ float lds_c[2][TKK][LDSP];   // 2*128*36*4 = 36864 B (55296 B)

    int tid  = threadIdx.x;
    int lane = tid & 31, w = tid >> 5;
    int lr = lane & 15, lg = lane >> 4;

    // wave w -> 32x32 output block: rows m_off..+31, cols k_off..+31
    int m_off = 32 * (w & 1);
    int k_off = 32 * (w >> 1);

    const float* Ab = data    + ((size_t)b * M_ + m0) * D_;
    const float* Cb = centers + ((size_t)b * K_ + k0) * D_;

    v8f acc00 = {}, acc01 = {}, acc10 = {}, acc11 = {};

    // async staging map: thread -> row (tid>>3, 0..31) and 16B column (tid&7)
    int srow = tid >> 3;
    int colF = (tid & 7) * 4;          // float column within 32-deep chunk
    // per-thread global byte offsets (depth term added per chunk)
    unsigned int gA0 = (unsigned int)(((srow      ) * D_ + colF) * 4);
    unsigned int gA1 = (unsigned int)(((srow + 32 ) * D_ + colF) * 4);
    unsigned int gC0 = (unsigned int)(((srow      ) * D_ + colF) * 4);
    unsigned int gC1 = (unsigned int)(((srow + 32 ) * D_ + colF) * 4);
    unsigned int gC2 = (unsigned int)(((srow + 64 ) * D_ + colF) * 4);
    unsigned int gC3 = (unsigned int)(((srow + 96 ) * D_ + colF) * 4);

    const int nchunks = D_ / DC;       // 24

    // issue async fills for buffer `buf` covering depth [dc, dc+DC)
    auto issue = [&](int buf, int dc) {
        unsigned int db = (unsigned int)(dc * 4);
        async_b128((unsigned int)(uintptr_t)&lds_a[buf][srow      ][colF], gA0 + db, Ab);
        async_b128((unsigned int)(uintptr_t)&lds_a[buf][srow + 32 ][colF], gA1 + db, Ab);
        async_b128((unsigned int)(uintptr_t)&lds_c[buf][srow      ][colF], gC0 + db, Cb);
        async_b128((unsigned int)(uintptr_t)&lds_c[buf][srow + 32 ][colF], gC1 + db, Cb);
        async_b128((unsigned int)(uintptr_t)&lds_c[buf][srow + 64 ][colF], gC2 + db, Cb);
        async_b128((unsigned int)(uintptr_t)&lds_c[buf][srow + 96 ][colF], gC3 + db, Cb);
    };

    issue(0, 0);
    wait_async0();
    __syncthreads();

    for (int c = 0; c < nchunks; ++c) {
        int cur = c & 1;
        if (c + 1 < nchunks) issue(cur ^ 1, (c + 1) * DC);

        // A 16x4 fp32 fragment: lane L -> row L%16, depth 2*(L/16)+j (j=0,1)
        // B 4x16 fragment (centers^T): same gather shape from lds_c
#pragma unroll
        for (int dd = 0; dd < DC; dd += 4) {
            v2f a0 = *(const v2f*)&lds_a[cur][m_off      + lr][dd + 2 * lg];
            v2f a1 = *(const v2f*)&lds_a[cur][m_off + 16 + lr][dd + 2 * lg];
            v2f b0 = *(const v2f*)&lds_c[cur][k_off      + lr][dd + 2 * lg];
            v2f b1 = *(const v2f*)&lds_c[cur][k_off + 16 + lr][dd + 2 * lg];
            acc00 = __builtin_amdgcn_wmma_f32_16x16x4_f32(
                false, a0, false, b0, (short)0, acc00, false, false);
            acc01 = __builtin_amdgcn_wmma_f32_16x16x4_f32(
                false, a0, false, b1, (short)0, acc01, false, false);
            acc10 = __builtin_amdgcn_wmma_f32_16x16x4_f32(
                false, a1, false, b0, (short)0, acc10, false, false);
            acc11 = __builtin_amdgcn_wmma_f32_16x16x4_f32(
                false, a1, false, b1, (short)0, acc11, false, false);
        }

        wait_async0();      // my next-buffer DMAs landed
        __syncthreads();    // everyone's landed; reads of `cur` finished
    }

    // Epilogue: D layout -> lane holds (m = base_m + v + 8*lg, k = base_k + lr).
    // score = ||c_k||^2 - 2*dot; pack (key32(score) << 32 | k); merge the two
    // k-subtiles in-register, min-reduce over the 16 lanes sharing the same m,
    // then atomicMin into the global per-row key.
    int kg0 = k0 + k_off + lr;
    int kg1 = kg0 + 16;
    float cn0 = cnorm[b * K_ + kg0];
    float cn1 = cnorm[b * K_ + kg1];
    unsigned long long* arow = assign + (size_t)b * M_;

#pragma unroll
    for (int ms = 0; ms < 2; ++ms) {
        v8f accL = ms ? acc10 : acc00;   // k-subtile 0
        v8f accR = ms ? acc11 : acc01;   // k-subtile 1
#pragma unroll
        for (int v = 0; v < 8; ++v) {
            float s0 = cn0 - 2.0f * accL[v];
            float s1 = cn1 - 2.0f * accR[v];
            unsigned long long key0 =
                ((unsigned long long)fkey(s0) << 32) | (unsigned)kg0;
            unsigned long long key1 =
                ((unsigned long long)fkey(s1) << 32) | (unsigned)kg1;
            unsigned long long key = key0 < key1 ? key0 : key1;
#pragma unroll
            for (int mask = 1; mask < 16; mask <<= 1) {
                unsigned long long o = shfl_xor_u64_16(key, mask);
                if (o < key) key = o;
            }
            if (lr == 0) {
                int mg = m0 + m_off + 16 * ms + v + 8 * lg;
                atomicMin(&arow[mg], key);
            }
        }
    }
}

// ---------------------------------------------------------------------------
// scatter: acc[k] += w_m * x_m, acc_counts[k] += w_m  (one block per row m)
// ---------------------------------------------------------------------------
__global__ void km_scatter_kernel(const float* __restrict__ data,
                                  const float* __restrict__ weights,
                                  const unsigned long long* __restrict__ assign,
                                  float* __restrict__ acc,
                                  float* __restrict__ acc_counts) {
    int b = blockIdx.y, m = blockIdx.x;                    // grid (1280, 4)
    unsigned long long key = assign[b * M_ + m];
    int k = (int)(unsigned int)(key & 0xffffffffull);
    float wv = weights[b * M_ + m];
    const float* src = data + ((size_t)b * M_ + m) * D_;
    float* dst = acc + ((size_t)b * K_ + k) * D_;
    if (threadIdx.x == 0) atomicAdd(&acc_counts[b * K_ + k], wv);
    for (int d = threadIdx.x; d < D_; d += blockDim.x)
        atomicAdd(&dst[d], wv * src[d]);
}

// ---------------------------------------------------------------------------
// finalize: centers = empty ? old : acc/(cnt+1e-8); counts = cnt
// ---------------------------------------------------------------------------
__global__ void km_finalize_kernel(const float* __restrict__ acc,
                                   const float* __restrict__ acc_counts,
                                   float* __restrict__ centers,
                                   float* __restrict__ counts) {
    int b = blockIdx.y, k = blockIdx.x;                    // grid (1024, 4)
    float cc = acc_counts[b * K_ + k];
    if (threadIdx.x == 0) counts[b * K_ + k] = cc;
    if (cc < 1e-6f) return;                                // empty: keep center
    float inv = 1.0f / (cc + 1e-8f);
    const float* a = acc + ((size_t)b * K_ + k) * D_;
    float* c = centers + ((size_t)b * K_ + k) * D_;
    for (int d = threadIdx.x; d < D_; d += blockDim.x) c[d] = a[d] * inv;
}

// ---------------------------------------------------------------------------
extern "C" void kernel_launch(void* const* d_in, const int* in_sizes, int n_in,
                              void* d_out, int out_size, void* d_ws, size_t ws_size,
                              hipStream_t stream) {
    const float* vf = (const float*)d_in[0];

    // workspace carve-up (~28.5 MB); d_out (B*K*D floats) doubles as the
    // scatter accumulator until the final D2D copy overwrites it.
    char* ws = (char*)d_ws;
    size_t off = 0;
    auto carve = [&](size_t bytes) -> void* {
        void* p = ws + off;
        off = (off + bytes + 255) & ~(size_t)255;
        return p;
    };
    float* centers    = (float*)carve((size_t)B_ * K_ * D_ * sizeof(float));
    float* data       = (float*)carve((size_t)B_ * M_ * D_ * sizeof(float));
    float* counts     = (float*)carve((size_t)B_ * K_ * sizeof(float));
    float* weights    = (float*)carve((size_t)B_ * M_ * sizeof(float));
    float* acc_counts = (float*)carve((size_t)B_ * K_ * sizeof(float));
    float* cnorm      = (float*)carve((size_t)B_ * K_ * sizeof(float));
    unsigned long long* assign =
        (unsigned long long*)carve((size_t)B_ * M_ * sizeof(unsigned long long));
    float* acc = (float*)d_out;

    dim3 blk(256);
    km_init_kernel<<<dim3(K_, B_), blk, 0, stream>>>(vf, centers, counts);

    for (int f = 0; f < NFRAMES; ++f) {
        km_build_data_kernel<<<dim3(M_, B_), blk, 0, stream>>>(
            vf, centers, counts, data, weights, f);
        for (int it = 0; it < NITERS; ++it) {
            km_prep_iter_kernel<<<dim3(M_, B_), blk, 0, stream>>>(
                centers, acc, acc_counts, cnorm, assign);
            km_score_argmin_kernel<<<dim3(M_ / TM, K_ / TKK, B_), blk, 0, stream>>>(
                data, centers, cnorm, assign);
            km_scatter_kernel<<<dim3(M_, B_), blk, 0, stream>>>(
                data, weights, assign, acc, acc_counts);
            km_finalize_kernel<<<dim3(K_, B_), blk, 0, stream>>>(
                acc, acc_counts, centers, counts);
        }
    }
    hipMemcpyAsync(d_out, centers, (size_t)B_ * K_ * D_ * sizeof(float),
                   hipMemcpyDeviceToDevice, stream);
}